// dct_layer_hy_12996571038115
// MI455X (gfx1250) — compile-verified
//
#include <hip/hip_runtime.h>
#include <hip/hip_bf16.h>

typedef __attribute__((ext_vector_type(2))) float v2f;
typedef __attribute__((ext_vector_type(8))) float v8f;

#define NDIM 1024
#define BM 128
#define BN 128
#define BK 32
#define APITCH 36    // floats; even (8B v2f reads), mult of 4 (16B async stores), conflict-free mod 64
#define BPITCH 128
#define NKT (NDIM / BK)

// ---------------------------------------------------------------------------
// Kernel 1: materialize the orthonormal DCT-II matrix D[m][k] into workspace.
//   D[m][k] = s_m * cos(pi*(2k+1)*m / 2048),  s_0 = 1/32, s_m = sqrt(2/1024)
// ---------------------------------------------------------------------------
__global__ __launch_bounds__(256) void dct_build_mat(float* __restrict__ Dm) {
    int idx = blockIdx.x * 256 + threadIdx.x;   // 0 .. 1024*1024-1
    int m = idx >> 10;
    int k = idx & 1023;
    int prod = ((2 * k + 1) * m) & 4095;        // exact (2k+1)*m mod 4096
    float ang = (float)prod * 1.5339807878856412e-3f;  // pi/2048
    float scale = (m == 0) ? 0.03125f                   // 1/sqrt(1024)
                           : 0.04419417382415922f;      // sqrt(2/1024)
    Dm[idx] = scale * cosf(ang);
}

// ---- CDNA5 async global->LDS copy (no VGPR staging; tracked by ASYNCcnt) ----
__device__ __forceinline__ void async_b128(unsigned lds_off, const float* g) {
    asm volatile("global_load_async_to_lds_b128 %0, %1, off"
                 :: "v"(lds_off), "v"(g) : "memory");
}
__device__ __forceinline__ void wait_async0() {
    asm volatile("s_wait_asynccnt 0x0" ::: "memory");
}

// ---------------------------------------------------------------------------
// Kernel 2: out[b] = D @ x[b]   (1024x1024x1024 SGEMM, 32 batches)
// 256 threads = 8 wave32s (4Mx2N wave grid), block tile 128x128, K-tile 32.
// Double-buffered LDS fed by async-to-LDS loads; one barrier per K-tile.
// ---------------------------------------------------------------------------
__global__ __launch_bounds__(256) void dct_gemm(const float* __restrict__ Dm,
                                                const float* __restrict__ X,
                                                float* __restrict__ Out) {
    __shared__ __align__(16) float As[2][BM * APITCH];   // 2 x 18432 B
    __shared__ __align__(16) float Bs[2][BK * BPITCH];   // 2 x 16384 B

    const int tid  = threadIdx.x;
    const int lane = tid & 31;
    const int wave = tid >> 5;
    const int wm   = wave & 3;      // wave M index: 0..3  (32 rows each)
    const int wn   = wave >> 2;     // wave N index: 0..1  (64 cols each)
    const int half = lane >> 4;     // lane half 0/1
    const int l16  = lane & 15;

    const int bm0   = blockIdx.y * BM;
    const int bn0   = blockIdx.x * BN;
    const int batch = blockIdx.z;
    const float* __restrict__ Xb = X   + (size_t)batch * NDIM * NDIM;
    float* __restrict__       Ob = Out + (size_t)batch * NDIM * NDIM;

    // Async-load indexing (16B per lane per op):
    //  A tile 128x32: 8 float4/row, 32 rows/pass, 4 passes.
    const int a_c4  = (tid & 7) * 4;
    const int a_row = tid >> 3;
    //  B tile 32x128: 32 float4/row, 8 rows/pass, 4 passes.
    const int b_c4  = (tid & 31) * 4;
    const int b_row = tid >> 5;

    auto issue_tile = [&](int kt, int bf) {
        const int k0 = kt * BK;
        #pragma unroll
        for (int p = 0; p < 4; ++p) {
            int r = a_row + p * 32;
            async_b128((unsigned)(size_t)&As[bf][r * APITCH + a_c4],
                       &Dm[(size_t)(bm0 + r) * NDIM + k0 + a_c4]);
        }
        #pragma unroll
        for (int p = 0; p < 4; ++p) {
            int r = b_row + p * 8;
            async_b128((unsigned)(size_t)&Bs[bf][r * BPITCH + b_c4],
                       &Xb[(size_t)(k0 + r) * NDIM + bn0 + b_c4]);
        }
    };

    v8f acc[2][4];
    {
        v8f zero = {};
        #pragma unroll
        for (int f = 0; f < 2; ++f)
            #pragma unroll
            for (int g = 0; g < 4; ++g)
                acc[f][g] = zero;
    }

    issue_tile(0, 0);   // prologue: tile 0 in flight

    for (int kt = 0; kt < NKT; ++kt) {
        const int cur = kt & 1;

        wait_async0();       // own tile-kt loads landed in LDS
        __syncthreads();     // everyone's landed; buf (1-cur) free of readers

        if (kt + 1 < NKT) issue_tile(kt + 1, 1 - cur);   // overlap with compute

        const float* __restrict__ Ac = As[cur];
        const float* __restrict__ Bc = Bs[cur];

        #pragma unroll
        for (int k4 = 0; k4 < BK / 4; ++k4) {
            const int acol = k4 * 4 + 2 * half;   // K pair this lane-half owns

            v2f afrag[2];
            #pragma unroll
            for (int f = 0; f < 2; ++f) {
                int m = wm * 32 + f * 16 + l16;
                afrag[f] = *(const v2f*)&Ac[m * APITCH + acol];
            }

            v2f bfrag[4];
            #pragma unroll
            for (int g = 0; g < 4; ++g) {
                int n = wn * 64 + g * 16 + l16;
                v2f b;
                b.x = Bc[(acol + 0) * BPITCH + n];
                b.y = Bc[(acol + 1) * BPITCH + n];
                bfrag[g] = b;
            }

            #pragma unroll
            for (int f = 0; f < 2; ++f)
                #pragma unroll
                for (int g = 0; g < 4; ++g)
                    acc[f][g] = __builtin_amdgcn_wmma_f32_16x16x4_f32(
                        /*neg_a=*/false, afrag[f],
                        /*neg_b=*/false, bfrag[g],
                        /*c_mod=*/(short)0, acc[f][g],
                        /*reuse_a=*/false, /*reuse_b=*/false);
        }
        // no trailing barrier: next iteration's top barrier covers the WAR
    }

    // ---- epilogue: C/D layout -> global. VGPR r: M = r + 8*half, N = l16 ----
    #pragma unroll
    for (int f = 0; f < 2; ++f) {
        #pragma unroll
        for (int g = 0; g < 4; ++g) {
            int col = bn0 + wn * 64 + g * 16 + l16;
            #pragma unroll
            for (int r = 0; r < 8; ++r) {
                int row = bm0 + wm * 32 + f * 16 + r + 8 * half;
                Ob[(size_t)row * NDIM + col] = acc[f][g][r];
            }
        }
    }
}

extern "C" void kernel_launch(void* const* d_in, const int* in_sizes, int n_in,
                              void* d_out, int out_size, void* d_ws, size_t ws_size,
                              hipStream_t stream) {
    const float* x = (const float*)d_in[0];
    float* out = (float*)d_out;
    float* Dm  = (float*)d_ws;   // needs 1024*1024*4 = 4 MB of scratch

    dct_build_mat<<<(NDIM * NDIM) / 256, 256, 0, stream>>>(Dm);

    dim3 grid(NDIM / BN, NDIM / BM, 32);   // 8 x 8 x 32 = 2048 blocks
    dct_gemm<<<grid, 256, 0, stream>>>(Dm, x, out);
}